// MultiAgentJSSPInitEmbedding_55181739819139
// MI455X (gfx1250) — compile-verified
//
#include <hip/hip_runtime.h>

typedef __attribute__((ext_vector_type(2))) float v2f;
typedef __attribute__((ext_vector_type(8))) float v8f;

#define B_ 64
#define J_ 128
#define O_ 64
#define M_ 32
#define E_ 128
#define POSROWS 127           // pos = next_op + o in [0, 126]
#define PE_STRIDE 132         // 132 % 8 == 4  ->  rows r and r+8 hit disjoint bank halves
#define INV_SCALE 0.01f

// ---------------------------------------------------------------------------
// Kernel 1: per-batch reductions (sched shift), and ma_emb output (1 MB).
// One block per batch b, 128 threads.
// ---------------------------------------------------------------------------
__global__ __launch_bounds__(128) void jssp_prep_kernel(
    const float* __restrict__ time_job_ready,  // [B,J]
    const float* __restrict__ time_ma_ready,   // [B,M]
    const int*   __restrict__ job_next_ma,     // [B,J]
    const float* __restrict__ w_ma,            // [E]
    float*       __restrict__ sched_ws,        // [B,J]  (workspace)
    float*       __restrict__ ma_out)          // [B,M,E]
{
    __shared__ float red[128];
    __shared__ float ma_shift[M_];
    const int b = blockIdx.x;
    const int t = threadIdx.x;

    // sched[b,j] = max(tjr, time_ma_ready[b, job_next_ma[b,j]])
    const float tjr  = time_job_ready[b * J_ + t];
    const int   mi   = job_next_ma[b * J_ + t];
    const float ama  = time_ma_ready[b * M_ + mi];
    const float sched = fmaxf(tjr, ama);

    red[t] = sched;
    __syncthreads();
    for (int s = 64; s > 0; s >>= 1) {
        if (t < s) red[t] = fminf(red[t], red[t + s]);
        __syncthreads();
    }
    const float smin = red[0];
    __syncthreads();
    sched_ws[b * J_ + t] = sched - smin;

    // ma_shift[m] = time_ma_ready[b,m] - min_m
    const float mv = (t < M_) ? time_ma_ready[b * M_ + t] : 3.4e38f;
    red[t] = mv;
    __syncthreads();
    for (int s = 64; s > 0; s >>= 1) {
        if (t < s) red[t] = fminf(red[t], red[t + s]);
        __syncthreads();
    }
    const float mmin = red[0];
    __syncthreads();
    if (t < M_) ma_shift[t] = mv - mmin;
    __syncthreads();

    // ma_emb[b,m,e] = ma_shift[m] * w_ma[e]; thread t == e, loop over m.
    const float wm = w_ma[t];
    #pragma unroll 4
    for (int m = 0; m < M_; ++m) {
        __builtin_nontemporal_store(ma_shift[m] * wm,
                                    &ma_out[(size_t)(b * M_ + m) * E_ + t]);
    }
}

// ---------------------------------------------------------------------------
// Kernel 2: ops_emb[b,j,o,e] via V_WMMA_F32_16X16X4_F32 tiles.
//   D(16o x 16e) = A(16o x 4) * B(4 x 16e) + C(PE tile from LDS)
//   A: k=0 -> proc/100, k=1 -> onehot(next_op)*sched/100, k=2,3 -> 0
//   B: k=0 -> w_ops[0], k=1 -> w_ops[1],  k=2,3 -> 0
// 256 blocks x 256 threads (8 waves). Block = (b, group of 32 j); each wave
// handles 4 j rows; PE table (127 x 132-stride) filled once per block.
// Linear padded layout => every C load is base-VGPR + immediate ds offset.
// ---------------------------------------------------------------------------
__global__ __launch_bounds__(256) void jssp_ops_kernel(
    const float* __restrict__ proc_times,  // [B,J,O]
    const int*   __restrict__ next_op,     // [B,J]
    const float* __restrict__ w_ops,       // [2,E]
    const float* __restrict__ sched_ws,    // [B,J]
    float*       __restrict__ ops_out)     // [B,J,O,E]
{
    __shared__ float pe[POSROWS * PE_STRIDE];   // 67,056 bytes

    const int tid   = threadIdx.x;
    const int lane  = tid & 31;
    const int wave  = tid >> 5;          // 0..7
    const int nle   = lane & 15;         // column/row index within half-wave
    const int rowg  = (lane >> 4) << 3;  // 0 or 8 (C/D row group per lane half)
    const int blk   = blockIdx.x;        // 0..255
    const int b     = blk >> 2;
    const int jbase = (blk & 3) * 32;

    // ---- fill PE table: pe[row*132 + e] = sin/cos(row * invf(e))
    const float NEG_LN1E4_OVER_E = -9.210340371976184f / 128.0f; // -ln(10000)/E
    for (int idx = tid; idx < POSROWS * E_; idx += 256) {
        const int row = idx >> 7;
        const int e   = idx & (E_ - 1);
        const float invf = __expf((float)(e & ~1) * NEG_LN1E4_OVER_E);
        const float ang  = (float)row * invf;
        const float v    = (e & 1) ? __cosf(ang) : __sinf(ang);
        pe[row * PE_STRIDE + e] = v;
    }

    // ---- B fragments for all 8 e-tiles (lanes 16..31 carry k=2,3 == 0)
    float bw0[8], bw1[8];
    #pragma unroll
    for (int et = 0; et < 8; ++et) {
        const float w0 = w_ops[et * 16 + nle];
        const float w1 = w_ops[E_ + et * 16 + nle];
        bw0[et] = (lane < 16) ? w0 : 0.0f;
        bw1[et] = (lane < 16) ? w1 : 0.0f;
    }
    __syncthreads();

    for (int ji = 0; ji < 4; ++ji) {
        const int bj = b * J_ + (jbase + ji * 8 + wave);
        const int nop = next_op[bj];
        const float sched = sched_ws[bj] * INV_SCALE;
        const float* __restrict__ prow = proc_times + (size_t)bj * O_;
        float* __restrict__ orow = ops_out + (size_t)bj * (O_ * E_)
                                 + rowg * E_ + nle;           // store base
        const float* __restrict__ cb = &pe[(nop + rowg) * PE_STRIDE + nle];

        // hoist the 4 proc loads (coalesced; halves mirror)
        float pv[4];
        #pragma unroll
        for (int om = 0; om < 4; ++om)
            pv[om] = prow[om * 16 + nle] * INV_SCALE;

        #pragma unroll
        for (int om = 0; om < 4; ++om) {
            const int o_l = om * 16 + nle;
            v2f A;
            A[0] = (lane < 16) ? pv[om] : 0.0f;
            A[1] = (lane < 16 && o_l == nop) ? sched : 0.0f;

            #pragma unroll
            for (int et = 0; et < 8; ++et) {
                v2f Bf;
                Bf[0] = bw0[et];
                Bf[1] = bw1[et];

                v8f C;
                #pragma unroll
                for (int r = 0; r < 8; ++r) {
                    // immediate ds offset: ((om*16+r)*132 + et*16)*4 <= 33,712 B
                    C[r] = cb[(om * 16 + r) * PE_STRIDE + et * 16];
                }

                // (neg_a, A, neg_b, B, c_mod, C, reuse_a, reuse_b)
                v8f D = __builtin_amdgcn_wmma_f32_16x16x4_f32(
                    false, A, false, Bf, (short)0, C, false, false);

                #pragma unroll
                for (int r = 0; r < 8; ++r) {
                    // immediate global offset: ((om*16+r)*128 + et*16)*4
                    __builtin_nontemporal_store(
                        D[r], &orow[(om * 16 + r) * E_ + et * 16]);
                }
            }
        }
    }
}

// ---------------------------------------------------------------------------
extern "C" void kernel_launch(void* const* d_in, const int* in_sizes, int n_in,
                              void* d_out, int out_size, void* d_ws, size_t ws_size,
                              hipStream_t stream) {
    const float* proc_times     = (const float*)d_in[0];  // [B,J,O]
    const float* time_job_ready = (const float*)d_in[1];  // [B,J]
    const float* time_ma_ready  = (const float*)d_in[2];  // [B,M]
    const int*   next_op        = (const int*)  d_in[3];  // [B,J]
    const int*   job_next_ma    = (const int*)  d_in[4];  // [B,J]
    const float* w_ops          = (const float*)d_in[5];  // [2,E]
    const float* w_ma           = (const float*)d_in[6];  // [1,E]

    float* ops_out = (float*)d_out;                              // [B,J,O,E]
    float* ma_out  = ops_out + (size_t)B_ * J_ * O_ * E_;        // [B,M,E]
    float* sched_ws = (float*)d_ws;                              // [B,J]

    jssp_prep_kernel<<<dim3(B_), dim3(J_), 0, stream>>>(
        time_job_ready, time_ma_ready, job_next_ma, w_ma, sched_ws, ma_out);

    jssp_ops_kernel<<<dim3(B_ * 4), dim3(256), 0, stream>>>(
        proc_times, next_op, w_ops, sched_ws, ops_out);
}